// Attention_81655918231876
// MI455X (gfx1250) — compile-verified
//
#include <hip/hip_runtime.h>

typedef float v2f __attribute__((ext_vector_type(2)));
typedef float v8f __attribute__((ext_vector_type(8)));

#define BATCH 2
#define HEADS 8
#define SEQ   2048
#define DH    64
#define DIM   512
#define QKV_N 1536
#define NEG_BIG (-1.0e30f)

static __device__ __forceinline__ v8f wmma_f32(v2f a, v2f b, v8f c) {
  // D = A(16x4 f32) * B(4x16 f32) + C(16x16 f32)
  return __builtin_amdgcn_wmma_f32_16x16x4_f32(false, a, false, b, (short)0, c,
                                               false, false);
}

static __device__ __forceinline__ v8f v8zero() {
  v8f z;
  for (int i = 0; i < 8; ++i) z[i] = 0.0f;
  return z;
}

// -------- CDNA5 async global->LDS copy (ASYNCcnt-tracked, §08 async_tensor) --
static __device__ __forceinline__ void async_b128(void* lds_dst,
                                                  const void* gsrc) {
  // VDST = 32-bit LDS byte address (low 32 bits of the generic pointer),
  // VADDR = 64-bit global address.
  unsigned int l = (unsigned int)(size_t)lds_dst;
  asm volatile("global_load_async_to_lds_b128 %0, %1, off"
               :
               : "v"(l), "v"(gsrc)
               : "memory");
}
static __device__ __forceinline__ void wait_async0() {
  asm volatile("s_wait_asynccnt 0x0" ::: "memory");
}

// ---------------------------------------------------------------------------
// Kernel 1: QKV projection.  x (B*SEQ x DIM) @ w_qkv (DIM x 1536),
// scattered into q/k/v workspaces laid out [b][h][n][64].
// ---------------------------------------------------------------------------
__global__ __launch_bounds__(256) void qkv_gemm(const float* __restrict__ x,
                                                const float* __restrict__ w,
                                                float* __restrict__ qw,
                                                float* __restrict__ kw,
                                                float* __restrict__ vw) {
  __shared__ __align__(16) float aT[16 * DIM];  // 32 KB A tile
  const int m0 = blockIdx.x * 16;
  const int tid = threadIdx.x;

  // async, coalesced staging of the 16 x 512 A tile
  {
    const float4* src = (const float4*)(x + (size_t)m0 * DIM);
    for (int idx = tid; idx < 16 * DIM / 4; idx += 256)
      async_b128(aT + idx * 4, src + idx);
    wait_async0();
  }
  __syncthreads();

  const int wave = tid >> 5;
  const int lane = tid & 31;
  const int l16 = lane & 15;
  const int hi = lane >> 4;
  const int n0 = blockIdx.y * 128 + wave * 16;

  v8f acc = v8zero();
  for (int kk = 0; kk < DIM / 4; ++kk) {
    const int kb = kk * 4 + hi * 2;
    v2f a = *(const v2f*)(aT + l16 * DIM + kb);  // A frag: M=l16, K=kb..kb+1
    v2f b;
    b.x = w[(size_t)(kb + 0) * QKV_N + n0 + l16];
    b.y = w[(size_t)(kb + 1) * QKV_N + n0 + l16];
    acc = wmma_f32(a, b, acc);
  }

  // scatter: column c in [0,1536): which = c/512, head = (c%512)/64, d = c%64
  const int c = n0 + l16;
  const int which = c / 512;
  const int h = (c % 512) / 64;
  const int d = c % 64;
  float* dst = (which == 0) ? qw : (which == 1 ? kw : vw);
  for (int r = 0; r < 8; ++r) {
    const int m = m0 + r + hi * 8;
    const int bb = m / SEQ;
    const int i = m % SEQ;
    dst[(((size_t)(bb * HEADS + h)) * SEQ + i) * DH + d] = acc[r];
  }
}

// ---------------------------------------------------------------------------
// Kernel 2: RoPE on K, and (scale * RoPE) on Q.  One wave per (b,h,i) row,
// lane handles dim-pair (2*lane, 2*lane+1).
// ---------------------------------------------------------------------------
__global__ __launch_bounds__(256) void rope_kernel(float* __restrict__ q,
                                                   float* __restrict__ k) {
  const int row = blockIdx.x * 8 + (threadIdx.x >> 5);  // [0, B*H*SEQ)
  const int lane = threadIdx.x & 31;
  const int pos = row % SEQ;
  const float inv_freq = __expf(-__logf(10000.0f) * (float)(2 * lane) / 64.0f);
  const float fr = (float)pos * inv_freq;
  float s, c;
  sincosf(fr, &s, &c);

  const size_t base = (size_t)row * DH + 2 * lane;
  const float k0 = k[base], k1 = k[base + 1];
  k[base]     = k0 * c - k1 * s;
  k[base + 1] = k1 * c + k0 * s;

  const float scale = 0.125f;  // DIM_HEAD^-0.5
  const float q0 = q[base] * scale, q1 = q[base + 1] * scale;
  q[base]     = q0 * c - q1 * s;
  q[base + 1] = q1 * c + q0 * s;
}

// ---------------------------------------------------------------------------
// Kernel 3: flash attention with additive pos_bias (bias fused as WMMA C).
// 4 waves / WG; each wave owns 16 query rows; K/V tiles are double-buffered
// in LDS and filled with global_load_async_to_lds_b128 (ASYNCcnt).
// ---------------------------------------------------------------------------
__global__ __launch_bounds__(128) void flash_attn(const float* __restrict__ q,
                                                  const float* __restrict__ k,
                                                  const float* __restrict__ v,
                                                  const float* __restrict__ bias,
                                                  float* __restrict__ outh) {
  __shared__ __align__(16) float kT[2][64 * DH];  // 2 x 16 KB
  __shared__ __align__(16) float vT[2][64 * DH];  // 2 x 16 KB
  __shared__ __align__(16) float pT[4][16 * DH];  // 16 KB, per-wave P staging

  const int bh = blockIdx.y;  // b*HEADS + h
  const int h = bh % HEADS;
  const int tid = threadIdx.x;
  const int wave = tid >> 5, lane = tid & 31;
  const int l16 = lane & 15, hi = lane >> 4;
  const int i0 = blockIdx.x * 64 + wave * 16;

  const float* qb = q + (size_t)bh * SEQ * DH;
  const float* kb = k + (size_t)bh * SEQ * DH;
  const float* vb = v + (size_t)bh * SEQ * DH;
  const float* biasb = bias + (size_t)h * SEQ * SEQ;

  // Q rows held in A-fragment registers for the whole kernel
  v2f aq[16];
  for (int kk = 0; kk < 16; ++kk)
    aq[kk] = *(const v2f*)(qb + (size_t)(i0 + l16) * DH + kk * 4 + hi * 2);

  float mrow[8], lrow[8];
  for (int r = 0; r < 8; ++r) { mrow[r] = NEG_BIG; lrow[r] = 0.0f; }
  v8f o[4];
  for (int dt = 0; dt < 4; ++dt) o[dt] = v8zero();

  // stage tile 0 into buffer 0 (async -> LDS, then drain ASYNCcnt + barrier)
  {
    const float4* ks = (const float4*)kb;
    const float4* vs = (const float4*)vb;
    for (int idx = tid; idx < 64 * DH / 4; idx += 128) {
      async_b128(&kT[0][idx * 4], ks + idx);
      async_b128(&vT[0][idx * 4], vs + idx);
    }
    wait_async0();
  }
  __syncthreads();

  const int NT = SEQ / 64;
  for (int jt = 0; jt < NT; ++jt) {
    const int j0 = jt * 64;
    const int cur = jt & 1;

    // prefetch next K/V tile into the other buffer (overlaps with compute)
    if (jt + 1 < NT) {
      const float4* ks = (const float4*)(kb + (size_t)(j0 + 64) * DH);
      const float4* vs = (const float4*)(vb + (size_t)(j0 + 64) * DH);
      float* kn = kT[1 - cur];
      float* vn = vT[1 - cur];
      for (int idx = tid; idx < 64 * DH / 4; idx += 128) {
        async_b128(kn + idx * 4, ks + idx);
        async_b128(vn + idx * 4, vs + idx);
      }
    }

    const float* kc = kT[cur];
    const float* vc = vT[cur];

    // S = Q @ K^T + bias   (streaming bias loaded NT straight into C operand)
    v8f s[4];
    for (int nt = 0; nt < 4; ++nt) {
      v8f c;
      for (int r = 0; r < 8; ++r)
        c[r] = __builtin_nontemporal_load(
            biasb + (size_t)(i0 + r + hi * 8) * SEQ + j0 + nt * 16 + l16);
      for (int kk = 0; kk < 16; ++kk) {
        // B frag of K^T: element (K=kd, N=jc) = K[jc][kd]
        v2f bf = *(const v2f*)(kc + (nt * 16 + l16) * DH + kk * 4 + hi * 2);
        c = wmma_f32(aq[kk], bf, c);
      }
      s[nt] = c;
    }

    // online softmax over the 64 new columns (16-lane row groups)
    for (int r = 0; r < 8; ++r) {
      float mx = fmaxf(fmaxf(s[0][r], s[1][r]), fmaxf(s[2][r], s[3][r]));
      for (int m = 1; m < 16; m <<= 1) mx = fmaxf(mx, __shfl_xor(mx, m, 32));
      const float mnew = fmaxf(mrow[r], mx);
      const float resc = __expf(mrow[r] - mnew);
      float sum = 0.0f;
      for (int nt = 0; nt < 4; ++nt) {
        const float p = __expf(s[nt][r] - mnew);
        s[nt][r] = p;
        sum += p;
      }
      for (int m = 1; m < 16; m <<= 1) sum += __shfl_xor(sum, m, 32);
      lrow[r] = lrow[r] * resc + sum;
      mrow[r] = mnew;
      for (int dt = 0; dt < 4; ++dt) o[dt][r] *= resc;
    }

    // C-layout -> A-layout for P via wave-private LDS slice
    for (int nt = 0; nt < 4; ++nt)
      for (int r = 0; r < 8; ++r)
        pT[wave][(r + hi * 8) * DH + nt * 16 + l16] = s[nt][r];

    // O += P @ V
    for (int kk = 0; kk < 16; ++kk) {
      v2f ap = *(const v2f*)(&pT[wave][l16 * DH + kk * 4 + hi * 2]);
      for (int dt = 0; dt < 4; ++dt) {
        v2f bv;
        bv.x = vc[(kk * 4 + hi * 2 + 0) * DH + dt * 16 + l16];
        bv.y = vc[(kk * 4 + hi * 2 + 1) * DH + dt * 16 + l16];
        o[dt] = wmma_f32(ap, bv, o[dt]);
      }
    }

    // drain this wave's async fills, then WG-wide handshake:
    // after this barrier the next tile is resident and the old one reusable.
    wait_async0();
    __syncthreads();
  }

  // epilogue: normalize and write heads back to [b][n][h*64+d]
  const int bb = bh / HEADS;
  for (int r = 0; r < 8; ++r) {
    const float inv = 1.0f / lrow[r];
    const int i = i0 + r + hi * 8;
    for (int dt = 0; dt < 4; ++dt)
      outh[((size_t)bb * SEQ + i) * (HEADS * DH) + h * DH + dt * 16 + l16] =
          o[dt][r] * inv;
  }
}

// ---------------------------------------------------------------------------
// Kernel 4: output projection.  out_heads (B*SEQ x 512) @ w_out (512 x 512).
// ---------------------------------------------------------------------------
__global__ __launch_bounds__(256) void out_gemm(const float* __restrict__ xin,
                                                const float* __restrict__ w,
                                                float* __restrict__ out) {
  __shared__ __align__(16) float aT[16 * DIM];
  const int m0 = blockIdx.x * 16;
  const int tid = threadIdx.x;
  {
    const float4* src = (const float4*)(xin + (size_t)m0 * DIM);
    for (int idx = tid; idx < 16 * DIM / 4; idx += 256)
      async_b128(aT + idx * 4, src + idx);
    wait_async0();
  }
  __syncthreads();

  const int wave = tid >> 5, lane = tid & 31;
  const int l16 = lane & 15, hi = lane >> 4;
  const int n0 = blockIdx.y * 128 + wave * 16;

  v8f acc = v8zero();
  for (int kk = 0; kk < DIM / 4; ++kk) {
    const int kb = kk * 4 + hi * 2;
    v2f a = *(const v2f*)(aT + l16 * DIM + kb);
    v2f b;
    b.x = w[(size_t)(kb + 0) * DIM + n0 + l16];
    b.y = w[(size_t)(kb + 1) * DIM + n0 + l16];
    acc = wmma_f32(a, b, acc);
  }
  for (int r = 0; r < 8; ++r) {
    const int m = m0 + r + hi * 8;
    out[(size_t)m * DIM + n0 + l16] = acc[r];
  }
}

// ---------------------------------------------------------------------------
extern "C" void kernel_launch(void* const* d_in, const int* in_sizes, int n_in,
                              void* d_out, int out_size, void* d_ws,
                              size_t ws_size, hipStream_t stream) {
  (void)in_sizes; (void)n_in; (void)out_size; (void)ws_size;
  const float* x        = (const float*)d_in[0];
  const float* pos_bias = (const float*)d_in[1];
  const float* w_qkv    = (const float*)d_in[2];
  const float* w_out    = (const float*)d_in[3];
  float* out = (float*)d_out;
  float* ws  = (float*)d_ws;

  const size_t per = (size_t)BATCH * HEADS * SEQ * DH;  // 2,097,152 floats
  float* qw = ws;
  float* kw = ws + per;
  float* vw = ws + 2 * per;
  float* oh = ws + 3 * per;  // [b][n][512] head-merged attention output

  dim3 g1(BATCH * SEQ / 16, QKV_N / 128);
  qkv_gemm<<<g1, 256, 0, stream>>>(x, w_qkv, qw, kw, vw);

  rope_kernel<<<BATCH * HEADS * SEQ / 8, 256, 0, stream>>>(qw, kw);

  dim3 g2(SEQ / 64, BATCH * HEADS);
  flash_attn<<<g2, 128, 0, stream>>>(qw, kw, vw, pos_bias, oh);

  dim3 g3(BATCH * SEQ / 16, DIM / 128);
  out_gemm<<<g3, 256, 0, stream>>>(oh, w_out, out);
}